// GSC_89653147336795
// MI455X (gfx1250) — compile-verified
//
#include <hip/hip_runtime.h>
#include <hip/hip_bf16.h>

typedef __attribute__((ext_vector_type(2))) float v2f;
typedef __attribute__((ext_vector_type(8))) float v8f;

#define GSC_B 8
#define GSC_T 600
#define GSC_F 257
#define GSC_C 8
#define GSC_M 7
#define GSC_TFULL 592   // last multiple-of-16 boundary below T handled unguarded

// ---------------------------------------------------------------------------
// Kernel 1: beamforming projections via V_WMMA_F32_16X16X4_F32.
// One wave per (b,f).  A = [16 time rows x 16 k] (k = {xr[0..7], xi[0..7]}),
// B = [16 k x 16 out] (out = {targ_r, targ_i, blk_r[0..6], blk_i[0..6]}),
// D accumulated over 4 chained K=4 WMMAs.  FP32 throughout (RLS downstream
// is numerically sensitive; op is bandwidth-bound so FP32 costs nothing).
// 37 full tiles store unconditionally; one epilogue tile has a single
// uniform half-wave predicate (rows 600..607 live only in lanes 16..31).
// ---------------------------------------------------------------------------
__global__ __launch_bounds__(32) void gsc_proj_wmma(
    const float* __restrict__ input,      // (B,T,2,F,C)
    const int*   __restrict__ beam_id,    // (B,)
    const float* __restrict__ targ_w_r,   // (NB,F,C)
    const float* __restrict__ targ_w_i,   // (NB,F,C)
    const float* __restrict__ block_w_r,  // (NB,M,F,C)
    const float* __restrict__ block_w_i,  // (NB,M,F,C)
    float*       __restrict__ proj)       // (B,T,F,16)
{
    const int f    = blockIdx.x;
    const int b    = blockIdx.y;
    const int lane = threadIdx.x;          // 0..31, full wave active at WMMA
    const int col  = lane & 15;            // N-column / A-row for this lane
    const int hi   = lane >> 4;            // 0: K+{0,1}, 1: K+{2,3}
    const int beam = beam_id[b];

    // Build this lane's B-matrix column: bcol[k] = B[k][col], k in 0..15.
    // k<8 -> coefficient of xr[c=k]; k>=8 -> coefficient of xi[c=k-8].
    const float* twr = targ_w_r + ((long)beam * GSC_F + f) * GSC_C;
    const float* twi = targ_w_i + ((long)beam * GSC_F + f) * GSC_C;
    float bcol[16];
    for (int c = 0; c < GSC_C; ++c) {
        float vr, vi;
        if (col == 0)       { vr =  twr[c];  vi = twi[c]; }   // targ_r
        else if (col == 1)  { vr = -twi[c];  vi = twr[c]; }   // targ_i
        else if (col < 9) {                                   // blk_r[m]
            const int m = col - 2;
            const long o = (((long)beam * GSC_M + m) * GSC_F + f) * GSC_C + c;
            vr = block_w_r[o]; vi = block_w_i[o];
        } else {                                              // blk_i[m]
            const int m = col - 9;
            const long o = (((long)beam * GSC_M + m) * GSC_F + f) * GSC_C + c;
            vr = -block_w_i[o]; vi = block_w_r[o];
        }
        bcol[c]     = vr;
        bcol[8 + c] = vi;
    }

    // B fragments (4x16 f32, 2 VGPRs): lanes 0-15 hold K={k0,k0+1},
    // lanes 16-31 hold K={k0+2,k0+3}, N = lane&15.
    const int koff = hi * 2;
    v2f bfrag[4];
    #pragma unroll
    for (int q = 0; q < 4; ++q) {
        bfrag[q].x = bcol[4 * q + koff + 0];
        bfrag[q].y = bcol[4 * q + koff + 1];
    }

    // Per-lane base pointers (A-row = col, time advances by 16 per tile).
    const float* aBase =
        input + (((long)b * GSC_T + col) * 2) * (long)GSC_F * GSC_C
              + (long)f * GSC_C;                       // + p*F*C + c per k
    float* sBase =
        proj + (((long)b * GSC_T + 8 * hi) * GSC_F + f) * 16 + col;

    const long aTileStride = 16L * 2 * GSC_F * GSC_C;  // 16 time rows
    const long sRowStride  = (long)GSC_F * 16;         // one time row
    const long pStride     = (long)GSC_F * GSC_C;      // xr -> xi plane

    // ---- 37 full tiles: completely unguarded hot loop ----
    for (int t0 = 0; t0 < GSC_TFULL; t0 += 16) {
        const float* ap = aBase + (long)t0 * 2 * GSC_F * GSC_C;
        v8f acc = {0.f, 0.f, 0.f, 0.f, 0.f, 0.f, 0.f, 0.f};
        #pragma unroll
        for (int q = 0; q < 4; ++q) {
            const int k0 = 4 * q + koff;               // k0, k0+1 contiguous c
            const int p  = k0 >> 3, c = k0 & 7;        // pair stays in one plane
            v2f a;
            a.x = ap[(long)p * pStride + c];
            a.y = ap[(long)p * pStride + c + 1];
            acc = __builtin_amdgcn_wmma_f32_16x16x4_f32(
                false, a, false, bfrag[q], (short)0, acc, false, false);
        }
        float* sp = sBase + (long)t0 * sRowStride;
        #pragma unroll
        for (int r = 0; r < 8; ++r)
            sp[(long)r * sRowStride] = acc[r];
    }

    // ---- epilogue tile t0 = 592: rows 592..607, clamp loads, store hi==0 ----
    {
        const int  t0 = GSC_TFULL;
        long t = t0 + col;
        if (t > GSC_T - 1) t = GSC_T - 1;
        v8f acc = {0.f, 0.f, 0.f, 0.f, 0.f, 0.f, 0.f, 0.f};
        #pragma unroll
        for (int q = 0; q < 4; ++q) {
            const int k0 = 4 * q + koff;
            const int p  = k0 >> 3, c = k0 & 7;
            v2f a;
            a.x = input[((((long)b * GSC_T + t) * 2 + p) * GSC_F + f) * GSC_C + c];
            a.y = input[((((long)b * GSC_T + t) * 2 + p) * GSC_F + f) * GSC_C + c + 1];
            acc = __builtin_amdgcn_wmma_f32_16x16x4_f32(
                false, a, false, bfrag[q], (short)0, acc, false, false);
        }
        if (hi == 0) {                                 // rows 592..599 only
            float* sp = sBase + (long)t0 * sRowStride;
            #pragma unroll
            for (int r = 0; r < 8; ++r)
                sp[(long)r * sRowStride] = acc[r];
        }
    }
}

// ---------------------------------------------------------------------------
// Kernel 2: per-(b,f) RLS recursion over t = 1..T-1.  One lane per (b,f);
// phi (7x7 complex) and W (7 complex) live in registers.  P@phi collapsed
// to rank-2 updates (O(M^2)).  Prefetches next step's projection row.
// ---------------------------------------------------------------------------
__global__ __launch_bounds__(256) void gsc_scan(
    const float* __restrict__ mask,   // (B,T,F)
    const float* __restrict__ proj,   // (B,T,F,16)
    float*       __restrict__ out)    // (B,T,2,F)
{
    const int idx = blockIdx.x * blockDim.x + threadIdx.x;
    if (idx >= GSC_B * GSC_F) return;
    const int b = idx / GSC_F;
    const int f = idx % GSC_F;

    float phr[GSC_M][GSC_M], phm[GSC_M][GSC_M];
    float Wr[GSC_M], Wi[GSC_M];
    #pragma unroll
    for (int i = 0; i < GSC_M; ++i) {
        #pragma unroll
        for (int j = 0; j < GSC_M; ++j) {
            phr[i][j] = (i == j) ? 100000.0f : 0.0f;
            phm[i][j] = 0.0f;
        }
        Wr[i] = 0.0f; Wi[i] = 0.0f;
    }

    // t = 0 slice is zero in the reference output.
    out[(((long)b * GSC_T + 0) * 2 + 0) * GSC_F + f] = 0.0f;
    out[(((long)b * GSC_T + 0) * 2 + 1) * GSC_F + f] = 0.0f;

    for (int t = 1; t < GSC_T; ++t) {
        const float* p = proj + (((long)b * GSC_T + t) * GSC_F + f) * 16;
        __builtin_prefetch(p + (long)GSC_F * 16, 0, 1);   // next time step

        const float m  = mask[((long)b * GSC_T + t) * GSC_F + f];
        const float tr = p[0];
        const float ti = p[1];
        float br[GSC_M], bi[GSC_M];
        #pragma unroll
        for (int i = 0; i < GSC_M; ++i) { br[i] = p[2 + i]; bi[i] = p[9 + i]; }

        float alpha = fminf(0.95f + 0.05f * m, 1.0f);
        if (m >= 0.5f) alpha = 1.0f;
        const float oma = 1.0f - alpha;

        // u = phi * bb  (complex matvec, row-major)
        float ur[GSC_M], ui[GSC_M];
        #pragma unroll
        for (int i = 0; i < GSC_M; ++i) {
            float sr = 0.0f, si = 0.0f;
            #pragma unroll
            for (int j = 0; j < GSC_M; ++j) {
                sr += phr[i][j] * br[j] - phm[i][j] * bi[j];
                si += phr[i][j] * bi[j] + phm[i][j] * br[j];
            }
            ur[i] = sr; ui[i] = si;
        }
        float low_r = 0.0f;
        #pragma unroll
        for (int i = 0; i < GSC_M; ++i) low_r += br[i] * ur[i] + bi[i] * ui[i];

        const float inv_den = 1.0f / (alpha + oma * low_r + 1e-13f);
        float kbr[GSC_M], kbi[GSC_M];
        #pragma unroll
        for (int i = 0; i < GSC_M; ++i) {
            kbr[i] = oma * ur[i] * inv_den;
            kbi[i] = oma * ui[i] * inv_den;
        }

        // v vectors: bb^T * phi  (columns)
        float vrr[GSC_M], vir[GSC_M], vri[GSC_M], vii[GSC_M];
        #pragma unroll
        for (int j = 0; j < GSC_M; ++j) {
            float a0 = 0.0f, a1 = 0.0f, a2 = 0.0f, a3 = 0.0f;
            #pragma unroll
            for (int k = 0; k < GSC_M; ++k) {
                a0 += br[k] * phr[k][j];
                a1 += bi[k] * phr[k][j];
                a2 += br[k] * phm[k][j];
                a3 += bi[k] * phm[k][j];
            }
            vrr[j] = a0; vir[j] = a1; vri[j] = a2; vii[j] = a3;
        }

        // phi = (phi - N) / (alpha + EPS);  N = rank-2 in kb
        const float ainv = 1.0f / (alpha + 1e-13f);
        #pragma unroll
        for (int i = 0; i < GSC_M; ++i) {
            #pragma unroll
            for (int j = 0; j < GSC_M; ++j) {
                const float Nr = kbr[i] * (vrr[j] + vii[j]) + kbi[i] * (vir[j] - vri[j]);
                const float Ni = kbr[i] * (vri[j] - vir[j]) + kbi[i] * (vii[j] + vrr[j]);
                phr[i][j] = (phr[i][j] - Nr) * ainv;
                phm[i][j] = (phm[i][j] - Ni) * ainv;
            }
        }

        // error & weight update
        float s_r = 0.0f, s_i = 0.0f;
        #pragma unroll
        for (int i = 0; i < GSC_M; ++i) {
            s_r += br[i] * Wr[i] + bi[i] * Wi[i];
            s_i += br[i] * Wi[i] - bi[i] * Wr[i];
        }
        const float e_r = tr - s_r;
        const float e_i = -ti - s_i;
        #pragma unroll
        for (int i = 0; i < GSC_M; ++i) {
            Wr[i] += kbr[i] * e_r - kbi[i] * e_i;
            Wi[i] += kbr[i] * e_i + kbi[i] * e_r;
        }

        // output
        float y_r = 0.0f, y_i = 0.0f;
        #pragma unroll
        for (int i = 0; i < GSC_M; ++i) {
            y_r += Wr[i] * br[i] + Wi[i] * bi[i];
            y_i += Wr[i] * bi[i] - Wi[i] * br[i];
        }
        out[(((long)b * GSC_T + t) * 2 + 0) * GSC_F + f] = m * (tr - y_r);
        out[(((long)b * GSC_T + t) * 2 + 1) * GSC_F + f] = m * (ti - y_i);
    }
}

extern "C" void kernel_launch(void* const* d_in, const int* in_sizes, int n_in,
                              void* d_out, int out_size, void* d_ws, size_t ws_size,
                              hipStream_t stream) {
    const float* input     = (const float*)d_in[0];
    const float* mask      = (const float*)d_in[1];
    const int*   beam_id   = (const int*)  d_in[2];
    const float* targ_w_r  = (const float*)d_in[3];
    const float* targ_w_i  = (const float*)d_in[4];
    const float* block_w_r = (const float*)d_in[5];
    const float* block_w_i = (const float*)d_in[6];
    float*       out       = (float*)d_out;
    float*       proj      = (float*)d_ws;   // (B,T,F,16) f32 = ~79 MB

    dim3 g1(GSC_F, GSC_B);
    gsc_proj_wmma<<<g1, 32, 0, stream>>>(input, beam_id, targ_w_r, targ_w_i,
                                         block_w_r, block_w_i, proj);

    const int nrec = GSC_B * GSC_F;
    gsc_scan<<<(nrec + 255) / 256, 256, 0, stream>>>(mask, proj, out);
}